// XFELMultiRes_59236188947237
// MI455X (gfx1250) — compile-verified
//
#include <hip/hip_runtime.h>

typedef float v2f __attribute__((ext_vector_type(2)));
typedef float v8f __attribute__((ext_vector_type(8)));

#define LDIM 128
#define INV127 (1.0f / 127.0f)

__device__ __forceinline__ int iclamp(int v, int lo, int hi) {
  return v < lo ? lo : (v > hi ? hi : v);
}

// One thread per output pixel. Each wave32 covers 32 consecutive k within one
// (b, j) row; the wave's 32 point rotations are done with two chained
// v_wmma_f32_16x16x4_f32 ops (rows 0-2 for lanes 0-15, rows 8-10 for lanes
// 16-31) so every lane reads (x,y,z) of its own point from acc[0..2].
__global__ __launch_bounds__(256) void xfel_project_kernel(
    const float* __restrict__ rotmat,
    const float* __restrict__ vol,
    float* __restrict__ out) {
  // 64 blocks of 256 threads per batch element (128*128 / 256 = 64)
  const int b      = blockIdx.x >> 6;
  const int pixoff = ((blockIdx.x & 63) << 8) + threadIdx.x;  // 0..16383
  const int j      = pixoff >> 7;          // plane row
  const int k      = pixoff & (LDIM - 1);  // plane col
  const int lane   = threadIdx.x & 31;
  const int m      = lane & 15;
  const bool hi    = lane >= 16;
  const int  k0    = k - lane;             // wave-uniform base column

  // ---- 3x3 rotation matrix: b is blockIdx-derived -> scalar (SMEM) loads ----
  const float* Rm = rotmat + b * 9;
  const float R00 = Rm[0], R01 = Rm[1], R02 = Rm[2];
  const float R10 = Rm[3], R11 = Rm[4], R12 = Rm[5];
  const float R20 = Rm[6], R21 = Rm[7], R22 = Rm[8];

  // A1[d,c] = R[c,d] in rows d=0..2 ; A2[8+d,c] = R[c,d] in rows 8..10.
  // 16x4 f32 A layout: lanes 0-15 hold K=0 (v.x), K=1 (v.y) for M=lane;
  //                    lanes 16-31 hold K=2 (v.x), K=3 (v.y) for M=lane-16.
  const float r0m = (m == 0) ? R00 : (m == 1) ? R01 : (m == 2) ? R02 : 0.0f;
  const float r1m = (m == 0) ? R10 : (m == 1) ? R11 : (m == 2) ? R12 : 0.0f;
  const float r2m = (m == 0) ? R20 : (m == 1) ? R21 : (m == 2) ? R22 : 0.0f;
  const float s0m = (m == 8) ? R00 : (m == 9) ? R01 : (m == 10) ? R02 : 0.0f;
  const float s1m = (m == 8) ? R10 : (m == 9) ? R11 : (m == 10) ? R12 : 0.0f;
  const float s2m = (m == 8) ? R20 : (m == 9) ? R21 : (m == 10) ? R22 : 0.0f;

  v2f a1, a2, b1, b2;
  a1.x = hi ? r2m : r0m;  a1.y = hi ? 0.0f : r1m;
  a2.x = hi ? s2m : s0m;  a2.y = hi ? 0.0f : s1m;

  // plane point n: ( lin[j], lin[64], lin[k0+n] ), lin[i] = -1 + 2i/127
  const float xj = fmaf((float)j, 2.0f * INV127, -1.0f);
  const float yc = INV127;  // lin[64]
  const float zA = fmaf((float)(k0 + m),      2.0f * INV127, -1.0f);
  const float zB = fmaf((float)(k0 + 16 + m), 2.0f * INV127, -1.0f);
  // B layout (4x16 f32): lanes 0-15 hold K=0 (v.x), K=1 (v.y) for N=lane;
  //                      lanes 16-31 hold K=2 (v.x), K=3 (v.y) for N=lane-16.
  b1.x = hi ? zA : xj;  b1.y = hi ? 0.0f : yc;
  b2.x = hi ? zB : xj;  b2.y = hi ? 0.0f : yc;

  v8f acc = {0.0f, 0.0f, 0.0f, 0.0f, 0.0f, 0.0f, 0.0f, 0.0f};
  acc = __builtin_amdgcn_wmma_f32_16x16x4_f32(false, a2, false, b2,
                                              (short)0, acc, false, false);
  acc = __builtin_amdgcn_wmma_f32_16x16x4_f32(false, a1, false, b1,
                                              (short)0, acc, false, false);

  // Every lane: acc[0..2] = rotated (x, y, z) of its own pixel's point.
  const float rx = acc[0];
  const float ry = acc[1];
  const float rz = acc[2];

  // ---- trilinear sample, align_corners=True, zero padding ----
  const float gx = (rx + 1.0f) * 63.5f;
  const float gy = (ry + 1.0f) * 63.5f;
  const float gz = (rz + 1.0f) * 63.5f;
  const float fx0 = floorf(gx), fy0 = floorf(gy), fz0 = floorf(gz);
  const float fx = gx - fx0, fy = gy - fy0, fz = gz - fz0;
  const int ix = (int)fx0, iy = (int)fy0, iz = (int)fz0;

  // fold per-axis validity into the weights (== w * where(valid, v, 0))
  const float wx0 = (ix     >= 0 && ix     < LDIM) ? (1.0f - fx) : 0.0f;
  const float wx1 = (ix + 1 >= 0 && ix + 1 < LDIM) ? fx          : 0.0f;
  const float wy0 = (iy     >= 0 && iy     < LDIM) ? (1.0f - fy) : 0.0f;
  const float wy1 = (iy + 1 >= 0 && iy + 1 < LDIM) ? fy          : 0.0f;
  const float wz0 = (iz     >= 0 && iz     < LDIM) ? (1.0f - fz) : 0.0f;
  const float wz1 = (iz + 1 >= 0 && iz + 1 < LDIM) ? fz          : 0.0f;

  const int cx0 = iclamp(ix, 0, LDIM - 1), cx1 = iclamp(ix + 1, 0, LDIM - 1);
  const int cy0 = iclamp(iy, 0, LDIM - 1), cy1 = iclamp(iy + 1, 0, LDIM - 1);
  const int cz0 = iclamp(iz, 0, LDIM - 1), cz1 = iclamp(iz + 1, 0, LDIM - 1);

  // vol is [D,H,W] = [z,y,x]; 8 MB -> resident in 192 MB L2, gathers are hits
  const int z0 = cz0 << 14, z1 = cz1 << 14;
  const int y0 = cy0 << 7,  y1 = cy1 << 7;
  const float v000 = vol[z0 + y0 + cx0];
  const float v001 = vol[z0 + y0 + cx1];
  const float v010 = vol[z0 + y1 + cx0];
  const float v011 = vol[z0 + y1 + cx1];
  const float v100 = vol[z1 + y0 + cx0];
  const float v101 = vol[z1 + y0 + cx1];
  const float v110 = vol[z1 + y1 + cx0];
  const float v111 = vol[z1 + y1 + cx1];

  const float sx00 = fmaf(v001, wx1, v000 * wx0);
  const float sx01 = fmaf(v011, wx1, v010 * wx0);
  const float sx10 = fmaf(v101, wx1, v100 * wx0);
  const float sx11 = fmaf(v111, wx1, v110 * wx0);
  const float sy0  = fmaf(sx01, wy1, sx00 * wy0);
  const float sy1  = fmaf(sx11, wy1, sx10 * wy0);
  float val        = fmaf(sy1,  wz1, sy0  * wz0);

  // ---- circular mask: ax[i] = (128*i - 8256)/127 ; radius 57.6 ----
  const float aj = (float)(128 * j - 8256) * INV127;
  const float ak = (float)(128 * k - 8256) * INV127;
  const bool inside = (aj * aj + ak * ak) < (57.6f * 57.6f);
  val = inside ? val : 0.0f;

  // streaming, write-once output: non-temporal store keeps L2 for the volume
  __builtin_nontemporal_store(val, &out[(b << 14) + pixoff]);
}

extern "C" void kernel_launch(void* const* d_in, const int* in_sizes, int n_in,
                              void* d_out, int out_size, void* d_ws, size_t ws_size,
                              hipStream_t stream) {
  const float* rotmat = (const float*)d_in[0];  // [B,3,3] f32
  const float* vol    = (const float*)d_in[1];  // [128,128,128] f32
  float* out          = (float*)d_out;          // [B,1,128,128] f32

  const int B = in_sizes[0] / 9;
  const int blocks = B * ((LDIM * LDIM) / 256);  // 64 blocks per batch element
  xfel_project_kernel<<<blocks, 256, 0, stream>>>(rotmat, vol, out);
}